// LineFinderLoss_66133906424051
// MI455X (gfx1250) — compile-verified
//
#include <hip/hip_runtime.h>
#include <hip/hip_bf16.h>

#define S 4096
#define ALPHA 0.01f
#define INV_S2 (1.0f / (4096.0f * 4096.0f))

typedef __attribute__((ext_vector_type(2))) float v2f;
typedef __attribute__((ext_vector_type(8))) float v8f;

// ---------------------------------------------------------------------------
// Workspace layout (floats):
//   [0 .. 4*S)   apack: float2[2*S]  WMMA A operand, pre-laid-out per lane-half
//                  apack[r]     = (-a0, -a1)              (lanes 0-15: K=0,1)
//                  apack[S + r] = (0.5*(a0^2+a1^2), 1.0)  (lanes 16-31: K=2,3)
//   [4*S .. 8*S) bpack: float2[2*S]  WMMA B operand
//                  bpack[m]     = (b0, b1)                (lanes 0-15: K=0,1)
//                  bpack[S + m] = (1.0, 0.5*(b0^2+b1^2))  (lanes 16-31: K=2,3)
//   [8*S ..)     partials[1024]      deterministic reduction staging
// ---------------------------------------------------------------------------

// De-stride the column gathers and materialize the exact WMMA lane layout so
// the cost kernel needs zero selects / zero divergence.
__global__ void __launch_bounds__(256) prep_kernel(const float* __restrict__ inp,
                                                   const float* __restrict__ tgt,
                                                   float* __restrict__ ws) {
    int r = blockIdx.x * blockDim.x + threadIdx.x;
    if (r < S) {
        v2f* apack = (v2f*)ws;
        v2f* bpack = (v2f*)(ws + 4 * S);
        float a0 = inp[(size_t)r * S + 0];
        float a1 = inp[(size_t)r * S + 1];
        v2f alo; alo.x = -a0;                      alo.y = -a1;
        v2f ahi; ahi.x = 0.5f * (a0*a0 + a1*a1);   ahi.y = 1.0f;
        apack[r]     = alo;
        apack[S + r] = ahi;
        float b0 = tgt[(size_t)r * S + 0];
        float b1 = tgt[(size_t)r * S + 1];
        v2f blo; blo.x = b0;   blo.y = b1;
        v2f bhi; bhi.x = 1.0f; bhi.y = 0.5f * (b0*b0 + b1*b1);
        bpack[r]     = blo;
        bpack[S + r] = bhi;
    }
}

// cost[n,m] = 0.5||a_n||^2 + 0.5||b_m||^2 - a_n.b_m via one
// V_WMMA_F32_16X16X4_F32 per 16x16 tile (K=4 padding trick):
//   A row: (-a0, -a1, 0.5||a||^2, 1)   B col: (b0, b1, 1, 0.5||b||^2)
// Inner loop: 1x global_load_b64 (B) + 1x wmma + 8x global_store_b32.
__global__ void __launch_bounds__(256) cost_wmma_kernel(const float* __restrict__ ws,
                                                        float* __restrict__ cost) {
    const v2f* apack = (const v2f*)ws;
    const v2f* bpack = (const v2f*)(ws + 4 * S);

    const int lane   = threadIdx.x & 31;
    const int wv     = threadIdx.x >> 5;      // 8 waves / block
    const int half   = lane >> 4;             // 0: K=0,1 lanes  1: K=2,3 lanes
    const int l15    = lane & 15;
    const int nTile  = blockIdx.x >> 3;       // 256 n-tiles
    const int mGroup = blockIdx.x & 7;        // 8 m-groups of 512 columns
    const int n0     = nTile * 16;
    const int hoff   = half * S;

    // A operand: one b64 load, fixed per wave, reused for 4 WMMAs.
    const v2f A = apack[hoff + n0 + l15];

    const v8f zero  = {};
    const int mBase = mGroup * 512 + wv * 64;

    #pragma unroll
    for (int t = 0; t < 4; ++t) {
        const int m = mBase + t * 16 + l15;
        const v2f B = bpack[hoff + m];

        v8f c = __builtin_amdgcn_wmma_f32_16x16x4_f32(
            /*neg_a=*/false, A, /*neg_b=*/false, B,
            /*c_mod=*/(short)0, zero, /*reuse_a=*/false, /*reuse_b=*/false);

        // D layout: VGPR v, lanes 0-15 -> row n0+v, lanes 16-31 -> row n0+8+v.
        const size_t rowBase = (size_t)(n0 + half * 8);
        const size_t col     = (size_t)m;
        #pragma unroll
        for (int v = 0; v < 8; ++v)
            cost[(rowBase + v) * S + col] = c[v];
    }
}

// Partial sums of ALPHA*(in-tg)^2 + INV_S2*(in[j,3]-tg[i,j])^2 over all S*S
// elements; one float partial per block, fixed-order tree => deterministic.
__global__ void __launch_bounds__(256) reduce_kernel(const float* __restrict__ inp,
                                                     const float* __restrict__ tgt,
                                                     float* __restrict__ partials) {
    __shared__ float col3[S];      // 16 KB: input[:,3] staged once per block
    __shared__ float sdata[256];

    for (int j = threadIdx.x; j < S; j += 256)
        col3[j] = inp[(size_t)j * S + 3];
    __syncthreads();

    const size_t total  = (size_t)S * S;
    const size_t stride = (size_t)gridDim.x * blockDim.x;
    float acc = 0.0f;
    for (size_t idx = (size_t)blockIdx.x * blockDim.x + threadIdx.x; idx < total; idx += stride) {
        const int   j = (int)(idx & (S - 1));
        const float t = tgt[idx];
        const float d = inp[idx] - t;
        const float e = col3[j] - t;
        acc = fmaf(ALPHA * d, d, acc);
        acc = fmaf(INV_S2 * e, e, acc);
    }

    sdata[threadIdx.x] = acc;
    __syncthreads();
    for (int s = 128; s > 0; s >>= 1) {
        if (threadIdx.x < (unsigned)s) sdata[threadIdx.x] += sdata[threadIdx.x + s];
        __syncthreads();
    }
    if (threadIdx.x == 0) partials[blockIdx.x] = sdata[0];
}

// Combine partials with the log-confidence term: loss = Sum(partials) - S*Sum(log c_n)
__global__ void __launch_bounds__(256) final_kernel(const float* __restrict__ inp,
                                                    const float* __restrict__ partials,
                                                    float* __restrict__ out) {
    __shared__ float sdata[256];
    float acc = 0.0f;
    for (int i = threadIdx.x; i < 1024; i += 256) acc += partials[i];
    float lg = 0.0f;
    for (int n = threadIdx.x; n < S; n += 256) lg += logf(inp[(size_t)n * S + 4]);
    sdata[threadIdx.x] = acc - (float)S * lg;
    __syncthreads();
    for (int s = 128; s > 0; s >>= 1) {
        if (threadIdx.x < (unsigned)s) sdata[threadIdx.x] += sdata[threadIdx.x + s];
        __syncthreads();
    }
    if (threadIdx.x == 0) out[0] = sdata[0];
}

extern "C" void kernel_launch(void* const* d_in, const int* in_sizes, int n_in,
                              void* d_out, int out_size, void* d_ws, size_t ws_size,
                              hipStream_t stream) {
    const float* inp = (const float*)d_in[0];   // input  [4096,4096] f32
    const float* tgt = (const float*)d_in[1];   // target [4096,4096] f32
    float* out = (float*)d_out;                 // [0]=loss, [1..]=cost row-major
    float* ws  = (float*)d_ws;
    float* partials = ws + 8 * S;

    prep_kernel     <<<(S + 255) / 256, 256, 0, stream>>>(inp, tgt, ws);
    cost_wmma_kernel<<<2048,            256, 0, stream>>>(ws, out + 1);
    reduce_kernel   <<<1024,            256, 0, stream>>>(inp, tgt, partials);
    final_kernel    <<<1,               256, 0, stream>>>(inp, partials, out);
}